// UFGConv_R_84018150244543
// MI455X (gfx1250) — compile-verified
//
#include <hip/hip_runtime.h>

#define NN   50000
#define NNZB 800000
#define FIN  128
#define FOUT 64
#define MT   (NN / 16)           // 3125 row tiles, exact
#define EDGE_THREADS (3 * NNZB * 16)   // 38,400,000 = 256 * 150000, exact

typedef float v2f __attribute__((ext_vector_type(2)));
typedef float v8f __attribute__((ext_vector_type(8)));

// ---------------------------------------------------------------------------
// Init: zero the three y buffers, broadcast bias into out (ws/out are poisoned)
// ---------------------------------------------------------------------------
__global__ __launch_bounds__(256) void ufg_init(float* __restrict__ y,
                                                float* __restrict__ out,
                                                const float* __restrict__ bias) {
    const int gid  = blockIdx.x * 256 + threadIdx.x;
    const int ytot = 3 * NN * FOUT;          // 9,600,000
    if (gid < ytot) {
        y[gid] = 0.0f;
    } else {
        const int o = gid - ytot;            // < NN*FOUT = 3,200,000
        out[o] = bias[o & (FOUT - 1)];
    }
}

// ---------------------------------------------------------------------------
// h = x @ W  via V_WMMA_F32_16X16X4_F32 (fp32 native matrix path).
// Block = 128 threads = 4 waves; each wave owns a 16x16 tile of the 16x64
// output stripe. x tile (16x128) and full W (128x64) staged in padded LDS.
// ---------------------------------------------------------------------------
__global__ __launch_bounds__(128) void ufg_gemm(const float* __restrict__ x,
                                                const float* __restrict__ w,
                                                float* __restrict__ h) {
    __shared__ float sx[16 * 132];   // row pad 128->132 (16B-aligned rows, conflict-free A reads)
    __shared__ float sw[128 * 68];   // row pad 64->68

    const int tid = threadIdx.x;
    const int m0  = blockIdx.x * 16;

    // Stage x tile: 16 rows x 128 cols, contiguous in global (512 float4)
    const float4* xg = (const float4*)(x + (size_t)m0 * FIN);
#pragma unroll
    for (int i = 0; i < 4; ++i) {
        const int g4 = tid + i * 128;        // 0..511
        const int r = g4 >> 5, c4 = g4 & 31;
        *(float4*)(&sx[r * 132 + c4 * 4]) = xg[g4];
    }
    // Stage full weight matrix: 128x64 (2048 float4)
    const float4* wg = (const float4*)w;
#pragma unroll
    for (int i = 0; i < 16; ++i) {
        const int g4 = tid + i * 128;        // 0..2047
        const int r = g4 >> 4, c4 = g4 & 15;
        *(float4*)(&sw[r * 68 + c4 * 4]) = wg[g4];
    }
    __syncthreads();

    const int wv   = tid >> 5;       // wave id 0..3 -> column tile
    const int lane = tid & 31;
    const int m    = lane & 15;      // A: M index / B,C,D: N index within tile
    const int hi   = lane >> 4;      // upper/lower half-wave
    const int n    = (wv << 4) + m;  // global output column 0..63

    v8f c = {};
#pragma unroll
    for (int kb = 0; kb < 32; ++kb) {
        const int k = kb * 4 + hi * 2;       // K base per half-wave (ISA 16x4 layout)
        v2f a, b;
        a.x = sx[m * 132 + k];
        a.y = sx[m * 132 + k + 1];
        b.x = sw[k * 68 + n];
        b.y = sw[(k + 1) * 68 + n];
        c = __builtin_amdgcn_wmma_f32_16x16x4_f32(
                /*neg_a=*/false, a, /*neg_b=*/false, b,
                /*c_mod=*/(short)0, c, /*reuse_a=*/false, /*reuse_b=*/false);
    }

    // C/D layout: VGPR i holds row (i + 8*hi), column = lane&15 (+tile offset)
    float* hp = h + (size_t)m0 * FOUT + n;
#pragma unroll
    for (int i = 0; i < 8; ++i) {
        hp[(size_t)(i + hi * 8) * FOUT] = c[i];
    }
}

// ---------------------------------------------------------------------------
// Pass 1: y[b][row] += val * h[col]   for b in {1,2,3}  (block 0 is dead)
// 16 threads per edge, float4 per thread (64 cols). Gathers hit L2 (h=12.8MB).
// ---------------------------------------------------------------------------
__global__ __launch_bounds__(256) void ufg_scatter1(const float* __restrict__ h,
                                                    const float* __restrict__ vals,
                                                    const int*   __restrict__ rows,
                                                    const int*   __restrict__ cols,
                                                    float*       __restrict__ y) {
    const int gid = blockIdx.x * 256 + threadIdx.x;        // < 3*NNZB*16 exact
    const int bi  = gid / (NNZB * 16);                     // 0..2 -> block bi+1
    const int rem = gid - bi * (NNZB * 16);
    const int e   = rem >> 4;
    const int r4  = (rem & 15) << 2;                       // column base 0..60

    const int idx = (bi + 1) * NNZB + e;
    const float v = vals[idx];
    const int row = rows[idx];
    const int col = cols[idx];

    const float4 hv = *(const float4*)(h + (size_t)col * FOUT + r4);
    float* yp = y + (size_t)bi * (NN * FOUT) + (size_t)row * FOUT + r4;
    __hip_atomic_fetch_add(yp + 0, v * hv.x, __ATOMIC_RELAXED, __HIP_MEMORY_SCOPE_AGENT);
    __hip_atomic_fetch_add(yp + 1, v * hv.y, __ATOMIC_RELAXED, __HIP_MEMORY_SCOPE_AGENT);
    __hip_atomic_fetch_add(yp + 2, v * hv.z, __ATOMIC_RELAXED, __HIP_MEMORY_SCOPE_AGENT);
    __hip_atomic_fetch_add(yp + 3, v * hv.w, __ATOMIC_RELAXED, __HIP_MEMORY_SCOPE_AGENT);
}

// ---------------------------------------------------------------------------
// Pass 2: out[row] += val * filt[b*N+col] * y[b][col]   (filt folded in here,
// saving a full 38MB read-modify-write sweep over y)
// ---------------------------------------------------------------------------
__global__ __launch_bounds__(256) void ufg_scatter2(const float* __restrict__ y,
                                                    const float* __restrict__ vals,
                                                    const int*   __restrict__ rows,
                                                    const int*   __restrict__ cols,
                                                    const float* __restrict__ filt,
                                                    float*       __restrict__ out) {
    const int gid = blockIdx.x * 256 + threadIdx.x;
    const int bi  = gid / (NNZB * 16);
    const int rem = gid - bi * (NNZB * 16);
    const int e   = rem >> 4;
    const int r4  = (rem & 15) << 2;

    const int idx = (bi + 1) * NNZB + e;
    const float v = vals[idx];
    const int row = rows[idx];
    const int col = cols[idx];

    const float s = v * filt[(size_t)(bi + 1) * NN + col];
    const float4 yv = *(const float4*)(y + (size_t)bi * (NN * FOUT) +
                                       (size_t)col * FOUT + r4);
    float* op = out + (size_t)row * FOUT + r4;
    __hip_atomic_fetch_add(op + 0, s * yv.x, __ATOMIC_RELAXED, __HIP_MEMORY_SCOPE_AGENT);
    __hip_atomic_fetch_add(op + 1, s * yv.y, __ATOMIC_RELAXED, __HIP_MEMORY_SCOPE_AGENT);
    __hip_atomic_fetch_add(op + 2, s * yv.z, __ATOMIC_RELAXED, __HIP_MEMORY_SCOPE_AGENT);
    __hip_atomic_fetch_add(op + 3, s * yv.w, __ATOMIC_RELAXED, __HIP_MEMORY_SCOPE_AGENT);
}

// ---------------------------------------------------------------------------
extern "C" void kernel_launch(void* const* d_in, const int* in_sizes, int n_in,
                              void* d_out, int out_size, void* d_ws, size_t ws_size,
                              hipStream_t stream) {
    const float* x    = (const float*)d_in[0];   // [N, FIN]
    const float* w    = (const float*)d_in[1];   // [FIN, FOUT]
    const float* filt = (const float*)d_in[2];   // [B*N, 1]
    const float* bias = (const float*)d_in[3];   // [FOUT]
    const float* vals = (const float*)d_in[4];   // [B, NNZB]
    const int*   rows = (const int*)d_in[5];     // [B, NNZB]
    const int*   cols = (const int*)d_in[6];     // [B, NNZB]
    float* out = (float*)d_out;                  // [N, FOUT]

    // Workspace layout: h [N*FOUT] | y [3*N*FOUT]  (total 51.2 MB)
    float* h = (float*)d_ws;
    float* y = h + (size_t)NN * FOUT;

    // init covers 3*N*FOUT (y) + N*FOUT (out) = 12,800,000 = 256*50000 exact
    ufg_init<<<50000, 256, 0, stream>>>(y, out, bias);
    ufg_gemm<<<MT, 128, 0, stream>>>(x, w, h);
    ufg_scatter1<<<EDGE_THREADS / 256, 256, 0, stream>>>(h, vals, rows, cols, y);
    ufg_scatter2<<<EDGE_THREADS / 256, 256, 0, stream>>>(y, vals, rows, cols, filt, out);
}